// DWT3DForward_66400194396245
// MI455X (gfx1250) — compile-verified
//
#include <hip/hip_runtime.h>

// ---------------------------------------------------------------------------
// 3D Haar DWT (J=2) for x:(4,1,128,256,256) f32.
// Each 2x2x2 input block -> 8 subband outputs, scale 1/(2*sqrt(2)).
// Channel order: c = 4*f_w + 2*f_h + f_d (f=1 is highpass). ll = c0; details c1..c7.
//
// CDNA5 path: per-block TDM tensor_load_to_lds of a (W, 2*Th, 2) f32 tile,
// s_wait_tensorcnt + workgroup barrier, butterfly in registers, NT stores
// for streaming detail subbands (keeps the 16MB LL1 intermediate hot in L2).
// ---------------------------------------------------------------------------

typedef unsigned int v4u __attribute__((ext_vector_type(4)));
typedef int          v4i __attribute__((ext_vector_type(4)));
typedef int          v8i __attribute__((ext_vector_type(8)));

#define HAAR_K 0.35355339059327379f  // 2^-1.5

__global__ void __launch_bounds__(512)
haar3d_level_kernel(const float* __restrict__ src,
                    int srcD, int srcH, int srcW,
                    float* __restrict__ ll_dst, long long llStrideN,
                    float* __restrict__ det_dst, long long detStrideC,
                    long long detStrideN,
                    int Th, int wshift)
{
    // max tile: 256 (W) * 8 (2*Th) * 2 (D) floats = 16 KB
    __shared__ __align__(16) float tile[4096];

    const int d_out  = blockIdx.x;   // output depth index (input planes 2d, 2d+1)
    const int h_tile = blockIdx.y;   // group of Th output rows
    const int n      = blockIdx.z;   // batch
    const int Wout   = srcW >> 1;
    const int Hout   = srcH >> 1;
    const int hrows  = Th << 1;      // input H rows in the tile

    // ---- TDM: DMA the (srcW x hrows x 2) f32 tile into LDS (wave 0 issues) ----
    if (threadIdx.x == 0) {
        unsigned lds_off = (unsigned)(size_t)(&tile[0]);   // low 32b = LDS byte offset
        unsigned long long elem =
            (((unsigned long long)(unsigned)(n * srcD + 2 * d_out) * (unsigned)srcH) +
             (unsigned long long)(unsigned)(h_tile * hrows)) * (unsigned)srcW;
        unsigned long long gaddr = (unsigned long long)(size_t)src + 4ull * elem;
        unsigned long long s0 = (unsigned)srcW;                                  // dim0 stride (elems)
        unsigned long long s1 = (unsigned long long)(unsigned)srcH * (unsigned)srcW; // dim1 stride

        // D# group 0: count=1(valid), lds_addr, global_addr[56:0], type=2
        v4u g0 = { 1u,
                   lds_off,
                   (unsigned)(gaddr & 0xFFFFFFFFull),
                   (unsigned)((gaddr >> 32) & 0x01FFFFFFull) | 0x80000000u };
        // D# group 1: mask=0, data_size=2 (4B), dims/strides/tile dims
        v8i g1 = {
            (int)(2u << 16),                                                        // [17:16] data_size=4B
            (int)(((unsigned)srcW & 0xFFFFu) << 16),                                // tensor_dim0[15:0]
            (int)((((unsigned)srcW >> 16) & 0xFFFFu) |
                  (((unsigned)srcH & 0xFFFFu) << 16)),                              // dim0 hi | dim1 lo
            (int)((((unsigned)srcH >> 16) & 0xFFFFu) |
                  (((unsigned)srcW & 0xFFFFu) << 16)),                              // dim1 hi | tile_dim0=srcW
            (int)(((unsigned)hrows & 0xFFFFu) | (2u << 16)),                        // tile_dim1=hrows, tile_dim2=2
            (int)(s0 & 0xFFFFFFFFull),                                              // dim0_stride lo32
            (int)(((s0 >> 32) & 0xFFFFull) | ((s1 & 0xFFFFull) << 16)),             // dim0_stride hi | dim1_stride lo16
            (int)((s1 >> 16) & 0xFFFFFFFFull)                                       // dim1_stride [47:16]
        };
        v4i g2 = { srcD, 0, 0, 0 };   // tensor_dim2 (group 2; rest unused/zero)
        v4i g3 = { 0, 0, 0, 0 };      // group 3 unused
        v8i gx = { 0, 0, 0, 0, 0, 0, 0, 0 };  // extra operand on clang-23 form
        __builtin_amdgcn_tensor_load_to_lds(g0, g1, g2, g3, gx, 0);
        __builtin_amdgcn_s_wait_tensorcnt(0);
    }
    __syncthreads();

    // ---- butterfly in registers ----
    const int w   = threadIdx.x & ((1 << wshift) - 1);  // w_out
    const int hl  = threadIdx.x >> wshift;              // local h_out (0..Th-1)
    const int hin = hl << 1;

    float2 p00 = *(const float2*)&tile[(0 * hrows + hin    ) * srcW + 2 * w]; // d=0,h=0
    float2 p01 = *(const float2*)&tile[(0 * hrows + hin + 1) * srcW + 2 * w]; // d=0,h=1
    float2 p10 = *(const float2*)&tile[(1 * hrows + hin    ) * srcW + 2 * w]; // d=1,h=0
    float2 p11 = *(const float2*)&tile[(1 * hrows + hin + 1) * srcW + 2 * w]; // d=1,h=1

    // W stage
    float l00 = p00.x + p00.y, h00 = p00.x - p00.y;
    float l01 = p01.x + p01.y, h01 = p01.x - p01.y;
    float l10 = p10.x + p10.y, h10 = p10.x - p10.y;
    float l11 = p11.x + p11.y, h11 = p11.x - p11.y;
    // H stage
    float ll0 = l00 + l01, lh0 = l00 - l01;
    float hl0 = h00 + h01, hh0 = h00 - h01;
    float ll1 = l10 + l11, lh1 = l10 - l11;
    float hl1 = h10 + h11, hh1 = h10 - h11;
    // D stage + scale; c = 4*f_w + 2*f_h + f_d
    float o0 = (ll0 + ll1) * HAAR_K;  // LLL (c0)
    float o1 = (ll0 - ll1) * HAAR_K;  // c1
    float o2 = (lh0 + lh1) * HAAR_K;  // c2
    float o3 = (lh0 - lh1) * HAAR_K;  // c3
    float o4 = (hl0 + hl1) * HAAR_K;  // c4
    float o5 = (hl0 - hl1) * HAAR_K;  // c5
    float o6 = (hh0 + hh1) * HAAR_K;  // c6
    float o7 = (hh0 - hh1) * HAAR_K;  // c7

    const int h_out = h_tile * Th + hl;
    const long long sp = ((long long)d_out * Hout + h_out) * Wout + w;

    ll_dst[(long long)n * llStrideN + sp] = o0;  // LL: regular store (re-read next level / final out)

    float* det = det_dst + (long long)n * detStrideN + sp;
    __builtin_nontemporal_store(o1, det + 0 * detStrideC);  // streaming: never re-read
    __builtin_nontemporal_store(o2, det + 1 * detStrideC);
    __builtin_nontemporal_store(o3, det + 2 * detStrideC);
    __builtin_nontemporal_store(o4, det + 3 * detStrideC);
    __builtin_nontemporal_store(o5, det + 4 * detStrideC);
    __builtin_nontemporal_store(o6, det + 5 * detStrideC);
    __builtin_nontemporal_store(o7, det + 6 * detStrideC);
}

extern "C" void kernel_launch(void* const* d_in, const int* in_sizes, int n_in,
                              void* d_out, int out_size, void* d_ws, size_t ws_size,
                              hipStream_t stream)
{
    (void)in_sizes; (void)n_in; (void)out_size; (void)ws_size;

    const float* x = (const float*)d_in[0];   // (4,1,128,256,256) f32
    float* out = (float*)d_out;
    float* ll1 = (float*)d_ws;                // (4,64,128,128) f32 = 16 MB scratch

    // d_out layout: [ll2: 4*32*64*64][yh1: 4*7*64*128*128][yh2: 4*7*32*64*64]
    float* ll2 = out;
    float* yh1 = out + (size_t)524288;
    float* yh2 = out + (size_t)524288 + (size_t)29360128;

    // Level 1: src (D,H,W)=(128,256,256) -> out spatial (64,128,128)
    // block = Wout*Th = 128*4 = 512 threads; grid = (Dout, Hout/Th, N)
    dim3 b1(512), g1(64, 32, 4);
    haar3d_level_kernel<<<g1, b1, 0, stream>>>(
        x, 128, 256, 256,
        ll1, 1048576LL,               // 64*128*128 per n
        yh1, 1048576LL, 7340032LL,    // chan stride, per-n stride (7 chans)
        /*Th=*/4, /*wshift=*/7);

    // Level 2: src (64,128,128) -> out spatial (32,64,64)
    dim3 b2(256), g2(32, 16, 4);
    haar3d_level_kernel<<<g2, b2, 0, stream>>>(
        ll1, 64, 128, 128,
        ll2, 131072LL,                // 32*64*64 per n
        yh2, 131072LL, 917504LL,
        /*Th=*/4, /*wshift=*/6);
}